// TTSA_Estimator_63333587747089
// MI455X (gfx1250) — compile-verified
//
#include <hip/hip_runtime.h>

#define NN 8192
#define AA 128
#define CC 100
#define MOM 0.3f
#define KCH 8              // samples staged per LDS chunk
#define LSTR 136           // padded LDS row stride in floats (2*136 % 64 == 16)

typedef __attribute__((ext_vector_type(2))) float v2f;
typedef __attribute__((ext_vector_type(8))) float v8f;

// ---------------------------------------------------------------------------
// Kernel 1: per-class counts, mean, and fused EMA vectors. Deterministic
// (fixed-order serial scan per class, no atomics).
// ---------------------------------------------------------------------------
__global__ __launch_bounds__(128) void stats_kernel(
    const float* __restrict__ feat, const int* __restrict__ labels,
    const float* __restrict__ em_ave, const float* __restrict__ ave_prev,
    float* __restrict__ ave, float* __restrict__ new_ave,
    float* __restrict__ new_em_ave, int* __restrict__ counts)
{
  const int c = blockIdx.x;     // class
  const int a = threadIdx.x;    // feature dim 0..127
  float s = 0.f;
  int cnt = 0;
  for (int i = 0; i < NN; ++i) {
    if (labels[i] == c) { s += feat[i * AA + a]; ++cnt; }
  }
  const float cs = (cnt == 0) ? 1.f : (float)cnt;
  const float av = s / cs;
  const int g = c * AA + a;
  ave[g]        = av;
  new_ave[g]    = MOM * ave_prev[g] + (1.f - MOM) * av;
  new_em_ave[g] = MOM * em_ave[g] + av;
  if (a == 0) counts[c] = cnt;
}

// ---------------------------------------------------------------------------
// Kernel 2: stable compaction of sample indices by class (wave32 ballot).
// One wave per class; deterministic ordering.
// ---------------------------------------------------------------------------
__global__ __launch_bounds__(32) void compact_kernel(
    const int* __restrict__ labels, const int* __restrict__ counts,
    int* __restrict__ offsets, int* __restrict__ sorted_idx)
{
  const int c = blockIdx.x;
  const int lane = threadIdx.x;     // 0..31 (wave32)
  int off = 0;
  for (int cc = 0; cc < c; ++cc) off += counts[cc];
  if (lane == 0) offsets[c] = off;
  int pos = off;
  for (int base = 0; base < NN; base += 32) {
    const int i = base + lane;
    const bool match = (labels[i] == c);
    const unsigned long long m = __ballot(match);
    if (match) {
      const int prefix = __popcll(m & ((1ull << lane) - 1ull));
      sorted_idx[pos + prefix] = i;
    }
    pos += __popcll(m);
  }
}

// ---------------------------------------------------------------------------
// Kernel 3: per-class centered Gram via V_WMMA_F32_16X16X4_F32 with LDS
// staging + fused covariance-update epilogue.
//
// One 1024-thread block (32 waves) per class. Block stages 8 centered sample
// rows per chunk into LDS (each class's rows touched in global exactly once);
// each wave owns 2 adjacent 16x16 tiles (shared A fragment, same row stripe)
// and feeds its WMMAs from ds_load.
//
// f32 A-matrix 16x4 layout (ISA 7.12.2): lanes 0-15 hold M=lane, K={0,1};
// lanes 16-31 hold M=lane-16, K={2,3}. B (4x16) mirrored.
// C/D: VGPR r -> M = r + 8*(lane>=16), N = lane&15.
// ---------------------------------------------------------------------------
__global__ __launch_bounds__(1024) void gram_cov_kernel(
    const float* __restrict__ feat,
    const float* __restrict__ em_covar,
    const float* __restrict__ em_ave_square,
    const float* __restrict__ init_cov,
    const float* __restrict__ init_ave,
    const int*   __restrict__ t_ptr,
    const float* __restrict__ ave_g,
    const float* __restrict__ new_ave_g,
    const float* __restrict__ new_em_ave_g,
    const int*   __restrict__ counts,
    const int*   __restrict__ offsets,
    const int*   __restrict__ sorted_idx,
    float* __restrict__ out)
{
  __shared__ float lds[2 * KCH * LSTR];   // double-buffered staging (8.7 KB)

  const int c    = blockIdx.x;          // class
  const int tid  = threadIdx.x;
  const int wave = tid >> 5;            // 0..31
  const int lane = tid & 31;
  const int half = lane >> 4;           // 0: supplies K {0,1}; 1: K {2,3}
  const int lm   = lane & 15;

  const int cnt = counts[c];
  const int off = offsets[c];
  const float* avec = ave_g + c * AA;

  // Two adjacent tiles per wave -> same row stripe tr, shared A fragment.
  const int tile0 = wave * 2;
  const int tr  = tile0 >> 3;
  const int tc0 = tile0 & 7;
  const int tc1 = tc0 + 1;

  const int acol  = tr  * 16 + lm;      // M feature dim (A operand)
  const int bcol0 = tc0 * 16 + lm;      // N feature dim, tile 0
  const int bcol1 = tc1 * 16 + lm;      // N feature dim, tile 1

  v8f acc0 = {}, acc1 = {};

  // Staging role: thread stages centered element (srow, scol) of each chunk.
  const int srow = tid >> 7;            // 0..7
  const int scol = tid & 127;           // 0..127
  const float s_ave = avec[scol];

  const int nchunks = (cnt + KCH - 1) / KCH;
  for (int ch = 0; ch < nchunks; ++ch) {
    float* buf = lds + (ch & 1) * (KCH * LSTR);
    {
      const int k = ch * KCH + srow;
      const float w = (k < cnt) ? 1.f : 0.f;          // zero-pad tail rows
      const int kc = (k < cnt) ? k : (cnt - 1);       // cnt >= 1 inside loop
      const int j = sorted_idx[off + kc];
      buf[srow * LSTR + scol] = w * (feat[j * AA + scol] - s_ave);
    }
    __syncthreads();   // single barrier per chunk: consume(ch-1) happened
                       // before stage(ch) in program order, so buffer reuse
                       // at ch+2 is already safe.
    for (int ks = 0; ks < KCH; ks += 4) {
      const int kA = ks + 2 * half;
      v2f aF, b0, b1;
      aF.x = buf[(kA)     * LSTR + acol];
      aF.y = buf[(kA + 1) * LSTR + acol];
      b0.x = buf[(kA)     * LSTR + bcol0];
      b0.y = buf[(kA + 1) * LSTR + bcol0];
      b1.x = buf[(kA)     * LSTR + bcol1];
      b1.y = buf[(kA + 1) * LSTR + bcol1];
      acc0 = __builtin_amdgcn_wmma_f32_16x16x4_f32(
          false, aF, false, b0, (short)0, acc0, false, false);
      acc1 = __builtin_amdgcn_wmma_f32_16x16x4_f32(
          false, aF, false, b1, (short)0, acc1, false, false);
    }
  }

  // Fused epilogue: new_cov = A1+A2+A3+A4+A5 - (A6 + A6^T), two tiles/wave.
  const float cs = (cnt == 0) ? 1.f : (float)cnt;
  const int t = t_ptr[0];
  float mt = 1.f;
  for (int j = 0; j < t; ++j) mt *= MOM;
  const float m = MOM, one_m = 1.f - MOM;

  const float* nav = new_ave_g    + c * AA;
  const float* nem = new_em_ave_g + c * AA;
  const float* iav = init_ave     + c * AA;

  for (int tile = 0; tile < 2; ++tile) {
    const int cb = (tile == 0 ? bcol0 : bcol1);
    const v8f acc = (tile == 0 ? acc0 : acc1);
    const float ave_b = avec[cb];
    const float nav_b = nav[cb];
    const float nem_b = nem[cb];
    const float iav_b = iav[cb];
    for (int r = 0; r < 8; ++r) {
      const int ra = tr * 16 + half * 8 + r;   // row index (M) for acc[r]
      const float ave_a = avec[ra];
      const float nav_a = nav[ra];
      const float nem_a = nem[ra];
      const float iav_a = iav[ra];
      const size_t idx = ((size_t)c * AA + ra) * AA + cb;
      const float vt = acc[r] / cs;            // var_temp element
      const float val =
            mt    * init_cov[idx]                               // A1
          + one_m * (m * em_covar[idx] + vt)                    // A2
          + mt    * iav_a * iav_b                               // A3
          + one_m * (m * em_ave_square[idx] + ave_a * ave_b)    // A4
          + nav_a * nav_b                                       // A5
          - mt    * (iav_a * nav_b + iav_b * nav_a)             // A6 + A6^T
          - one_m * (nem_a * nav_b + nem_b * nav_a);            // A6 + A6^T
      out[idx] = val;
    }
  }
}

// ---------------------------------------------------------------------------
extern "C" void kernel_launch(void* const* d_in, const int* in_sizes, int n_in,
                              void* d_out, int out_size, void* d_ws, size_t ws_size,
                              hipStream_t stream) {
  const float* feat     = (const float*)d_in[0];
  const float* em_covar = (const float*)d_in[1];
  const float* em_avesq = (const float*)d_in[2];
  const float* em_ave   = (const float*)d_in[3];
  const float* ave_prev = (const float*)d_in[4];
  const float* init_cov = (const float*)d_in[5];
  const float* init_ave = (const float*)d_in[6];
  const int*   labels   = (const int*)d_in[7];
  const int*   t_ptr    = (const int*)d_in[8];
  float* out = (float*)d_out;

  char* ws = (char*)d_ws;
  float* ave        = (float*)ws; ws += (size_t)CC * AA * sizeof(float);
  float* new_ave    = (float*)ws; ws += (size_t)CC * AA * sizeof(float);
  float* new_em_ave = (float*)ws; ws += (size_t)CC * AA * sizeof(float);
  int*   counts     = (int*)ws;   ws += (size_t)CC * sizeof(int);
  int*   offsets    = (int*)ws;   ws += (size_t)CC * sizeof(int);
  int*   sorted_idx = (int*)ws;   ws += (size_t)NN * sizeof(int);

  stats_kernel<<<CC, AA, 0, stream>>>(feat, labels, em_ave, ave_prev,
                                      ave, new_ave, new_em_ave, counts);
  compact_kernel<<<CC, 32, 0, stream>>>(labels, counts, offsets, sorted_idx);
  gram_cov_kernel<<<CC, 1024, 0, stream>>>(feat, em_covar, em_avesq, init_cov,
                                           init_ave, t_ptr, ave, new_ave,
                                           new_em_ave, counts, offsets,
                                           sorted_idx, out);
}